// EmbeddingGrad_48137993454109
// MI455X (gfx1250) — compile-verified
//
#include <hip/hip_runtime.h>
#include <hip/hip_bf16.h>
#include <stdint.h>

// Problem constants (from the reference)
#define NUM_EMBEDDING 8192   // V
#define EMBEDDING_DIM 128    // D
#define BATCH 1024
#define HIST_LEN 50
#define NPAIRS (BATCH * HIST_LEN)   // 51200 lookups

#define WAVES_PER_BLOCK 8
#define PAIRS_PER_WAVE 4
#define PAIRS_PER_BLOCK (WAVES_PER_BLOCK * PAIRS_PER_WAVE)  // 32

// ---------------------------------------------------------------------------
// Kernel 1: Wt[v][d] = W[d][v] + b[d]
// Standard 32x32 LDS-tiled transpose (coalesced read along v, coalesced write
// along d), with the bias folded in so the gather is a pure row copy.
// Runs once over 4 MB -> negligible vs. the gather.
// ---------------------------------------------------------------------------
__global__ void __launch_bounds__(256)
transpose_bias_kernel(const float* __restrict__ W,
                      const float* __restrict__ b,
                      float* __restrict__ Wt) {
    __shared__ float tile[32][33];   // +1 pad: no LDS bank conflicts

    const int v0 = blockIdx.x * 32;  // vocab tile origin
    const int d0 = blockIdx.y * 32;  // dim tile origin
    const int tx = threadIdx.x;      // 0..31
    const int ty = threadIdx.y;      // 0..7

    // Read: each wave reads one row of W (contiguous in v)
    #pragma unroll
    for (int i = 0; i < 32; i += 8) {
        const int d = d0 + ty + i;
        tile[ty + i][tx] = W[(size_t)d * NUM_EMBEDDING + (v0 + tx)];
    }
    __syncthreads();

    // Write: each wave writes one row of Wt (contiguous in d), bias folded in
    const float bias = b[d0 + tx];
    #pragma unroll
    for (int i = 0; i < 32; i += 8) {
        const int v = v0 + ty + i;
        Wt[(size_t)v * EMBEDDING_DIM + (d0 + tx)] = tile[tx][ty + i] + bias;
    }
}

// ---------------------------------------------------------------------------
// Kernel 2: out[p][:] = Wt[idx[p]][:]  via the CDNA5 async global<->LDS path.
// One wave per 4 lookups; each lane moves 16 B. Loads are issued async into a
// per-wave LDS staging buffer (ASYNCcnt), then stored async from LDS to the
// output row -- the data never transits VGPRs. 4-deep software pipeline per
// wave; a final s_wait_asynccnt 0 covers the final stores.
// ---------------------------------------------------------------------------
__global__ void __launch_bounds__(256)
gather_async_kernel(const int* __restrict__ idx,
                    const float* __restrict__ Wt,
                    float* __restrict__ out,
                    int npairs) {
    __shared__ float4 stage[WAVES_PER_BLOCK][PAIRS_PER_WAVE][32];

    const int wave = threadIdx.x >> 5;
    const int lane = threadIdx.x & 31;
    const int base = (blockIdx.x * WAVES_PER_BLOCK + wave) * PAIRS_PER_WAVE;

    // LDS byte address: low 32 bits of the generic pointer are the LDS offset.
    const uint32_t lds0 = (uint32_t)(uintptr_t)(&stage[wave][0][lane]);

    // Issue all async gathers (memory -> LDS), one b128 per lane per row.
    #pragma unroll
    for (int p = 0; p < PAIRS_PER_WAVE; ++p) {
        if (base + p < npairs) {                       // wave-uniform guard
            const int id = idx[base + p];
            const uint64_t src =
                (uint64_t)(uintptr_t)(Wt + (size_t)id * EMBEDDING_DIM + lane * 4);
            const uint32_t la = lds0 + (uint32_t)(p * 32 * sizeof(float4));
            asm volatile("global_load_async_to_lds_b128 %0, %1, off"
                         :: "v"(la), "v"(src) : "memory");
        }
    }

    // One wait for all outstanding async loads of this wave.
    asm volatile("s_wait_asynccnt 0x0" ::: "memory");

    // Issue all async stores (LDS -> output rows).
    #pragma unroll
    for (int p = 0; p < PAIRS_PER_WAVE; ++p) {
        if (base + p < npairs) {
            const uint64_t dst =
                (uint64_t)(uintptr_t)(out + (size_t)(base + p) * EMBEDDING_DIM + lane * 4);
            const uint32_t la = lds0 + (uint32_t)(p * 32 * sizeof(float4));
            asm volatile("global_store_async_from_lds_b128 %0, %1, off"
                         :: "v"(dst), "v"(la) : "memory");
        }
    }
    asm volatile("s_wait_asynccnt 0x0" ::: "memory");
}

// ---------------------------------------------------------------------------
// Fallback gather (no workspace): direct strided column read of W + bias.
// Only used if ws_size can't hold the 4 MB transposed table.
// ---------------------------------------------------------------------------
__global__ void __launch_bounds__(256)
gather_direct_kernel(const int* __restrict__ idx,
                     const float* __restrict__ W,
                     const float* __restrict__ b,
                     float* __restrict__ out,
                     int npairs) {
    const int pair = blockIdx.x * 2 + (threadIdx.x >> 7);
    const int d = threadIdx.x & 127;
    if (pair < npairs) {
        const int id = idx[pair];
        out[(size_t)pair * EMBEDDING_DIM + d] =
            W[(size_t)d * NUM_EMBEDDING + id] + b[d];
    }
}

extern "C" void kernel_launch(void* const* d_in, const int* in_sizes, int n_in,
                              void* d_out, int out_size, void* d_ws, size_t ws_size,
                              hipStream_t stream) {
    const int*   idx = (const int*)d_in[0];   // [1024, 50] int32
    const float* W   = (const float*)d_in[1]; // [128, 8192] f32
    const float* b   = (const float*)d_in[2]; // [128] f32
    float*       out = (float*)d_out;         // [1024, 50, 128] f32

    const int npairs = in_sizes[0];           // 51200
    const size_t wt_bytes = (size_t)NUM_EMBEDDING * EMBEDDING_DIM * sizeof(float); // 4 MB

    if (ws_size >= wt_bytes) {
        float* Wt = (float*)d_ws;

        // 1) Transpose + bias fold: W[128,8192] -> Wt[8192,128]
        dim3 tb(32, 8);
        dim3 tg(NUM_EMBEDDING / 32, EMBEDDING_DIM / 32);   // (256, 4)
        transpose_bias_kernel<<<tg, tb, 0, stream>>>(W, b, Wt);

        // 2) Async row gather: 32 lookups per 256-thread block
        const int blocks = (npairs + PAIRS_PER_BLOCK - 1) / PAIRS_PER_BLOCK; // 1600
        gather_async_kernel<<<blocks, 256, 0, stream>>>(idx, Wt, out, npairs);
    } else {
        const int blocks = (npairs + 1) / 2;
        gather_direct_kernel<<<blocks, 256, 0, stream>>>(idx, W, b, out, npairs);
    }
}